// SelfAttention_74698071212229
// MI455X (gfx1250) — compile-verified
//
#include <hip/hip_runtime.h>

typedef float v2f __attribute__((ext_vector_type(2)));
typedef float v8f __attribute__((ext_vector_type(8)));

#define WMMA_F32(a, b, c) \
  __builtin_amdgcn_wmma_f32_16x16x4_f32(false, (a), false, (b), (short)0, (c), false, false)

// ---------------------------------------------------------------------------
// GEMM: C[M,N] = A[M,K] @ B + bias[N]
//   wmode 0: B row-major [K][N]                     (W_O path)
//   wmode 1: B is W_{Q,K,V} [H][K][64]: B[k][n] = W[(n>>6)*K*64 + k*64 + (n&63)]
//   omode 0: C row-major [M][N]
//   omode 1: C k-paired per (b,h):  C[(b*16+h)][s>>1][e][s&1]  (V for z-phase)
// Block: 256 threads (8 waves), tile 128x64, kc=16, LDS double-buffered,
// one barrier per stage; B tile stored k-paired so every B fragment is a
// single aligned ds_load_b64 into an adjacent VGPR pair (no packing movs).
// ---------------------------------------------------------------------------
#define BPS 160   // B pair-row stride (floats); 160 % 64 == 32 -> halves disjoint

__global__ __launch_bounds__(256) void gemm_wmma_f32(
    const float* __restrict__ A, const float* __restrict__ Bw,
    const float* __restrict__ bias, float* __restrict__ C,
    int M, int N, int K, int wmode, int omode)
{
  __shared__ float As[2][128 * 18];  // padded stride 18: conflict-free A-frag reads
  __shared__ float Bs[2][8 * BPS];   // k-paired: B[k][n] at (k>>1)*BPS + n*2 + (k&1)

  const int nTiles = N >> 6;
  const int m0 = (blockIdx.x / nTiles) << 7;
  const int n0 = (blockIdx.x % nTiles) << 6;
  const int t    = threadIdx.x;
  const int wave = t >> 5;
  const int lane = t & 31;
  const int ml   = lane & 15;        // M (A/C) or N (B) index within tile
  const int hi   = lane >> 4;        // half-wave select

  // A staging: 2 float4 per thread per stage
  const int aRow0 = (t << 2) >> 4;         // 0..63
  const int aRow1 = aRow0 + 64;            // 64..127
  const int aCol  = (t << 2) & 15;
  // B staging: one k-row pair, two columns per thread (2 x float2 loads)
  const int pr = t >> 5;                   // 0..7  (k rows 2pr, 2pr+1)
  const int pc = (lane) << 1;              // 0..62 (even column)

  const float* Ap0 = A + (size_t)(m0 + aRow0) * K + aCol;
  const float* Ap1 = A + (size_t)(m0 + aRow1) * K + aCol;
  const float* Bp0 = (wmode == 0)
      ? Bw + (size_t)(2 * pr) * N + n0 + pc
      : Bw + (size_t)(n0 >> 6) * K * 64 + (size_t)(2 * pr) * 64 + pc;
  const float* Bp1 = Bp0 + ((wmode == 0) ? N : 64);
  const size_t bStep = (wmode == 0) ? (size_t)16 * N : (size_t)16 * 64;

  v8f acc[4] = {};

  // ---- prologue: stage K-tile 0 into buffer 0 ----
  {
    const float4 ra0 = *(const float4*)(Ap0);
    const float4 ra1 = *(const float4*)(Ap1);
    const float2 rb0 = *(const float2*)(Bp0);
    const float2 rb1 = *(const float2*)(Bp1);
    float* d0 = &As[0][aRow0 * 18 + aCol];
    d0[0] = ra0.x; d0[1] = ra0.y; d0[2] = ra0.z; d0[3] = ra0.w;
    float* d1 = &As[0][aRow1 * 18 + aCol];
    d1[0] = ra1.x; d1[1] = ra1.y; d1[2] = ra1.z; d1[3] = ra1.w;
    float4 st; st.x = rb0.x; st.y = rb1.x; st.z = rb0.y; st.w = rb1.y;
    *(float4*)&Bs[0][pr * BPS + pc * 2] = st;
  }
  __syncthreads();

  int buf = 0;
  size_t bOff = bStep;
  for (int k0 = 0; k0 < K; k0 += 16) {
    const bool hasNext = (k0 + 16) < K;
    float4 ra0, ra1;
    float2 rb0, rb1;
    if (hasNext) {                       // issue loads early: latency hidden by WMMAs
      ra0 = *(const float4*)(Ap0 + k0 + 16);
      ra1 = *(const float4*)(Ap1 + k0 + 16);
      rb0 = *(const float2*)(Bp0 + bOff);
      rb1 = *(const float2*)(Bp1 + bOff);
    }

    #pragma unroll
    for (int kk = 0; kk < 16; kk += 4) {
      const int kb = kk + hi * 2;        // even
      const int kp = kb >> 1;            // pair row 0..7
      const v2f a = *(const v2f*)&As[buf][(wave * 16 + ml) * 18 + kb];
      #pragma unroll
      for (int ns = 0; ns < 4; ns++) {
        const v2f b = *(const v2f*)&Bs[buf][kp * BPS + (ns * 16 + ml) * 2];
        acc[ns] = WMMA_F32(a, b, acc[ns]);
      }
    }

    if (hasNext) {                       // fill the other buffer, single barrier
      float* d0 = &As[buf ^ 1][aRow0 * 18 + aCol];
      d0[0] = ra0.x; d0[1] = ra0.y; d0[2] = ra0.z; d0[3] = ra0.w;
      float* d1 = &As[buf ^ 1][aRow1 * 18 + aCol];
      d1[0] = ra1.x; d1[1] = ra1.y; d1[2] = ra1.z; d1[3] = ra1.w;
      float4 st; st.x = rb0.x; st.y = rb1.x; st.z = rb0.y; st.w = rb1.y;
      *(float4*)&Bs[buf ^ 1][pr * BPS + pc * 2] = st;
      __syncthreads();
      bOff += bStep;
    }
    buf ^= 1;
  }

  #pragma unroll
  for (int ns = 0; ns < 4; ns++) {
    const int col = n0 + ns * 16 + ml;
    const float bv = bias[col];
    #pragma unroll
    for (int r = 0; r < 8; r++) {
      const int row = m0 + wave * 16 + r + hi * 8;
      const float val = acc[ns][r] + bv;
      if (omode == 0) {
        C[(size_t)row * N + col] = val;
      } else {
        // V k-paired: per (b,h) region of 1024*128 floats
        const int s = row & 2047, bb = row >> 11;
        const int hh = col >> 6, e = col & 63;
        C[((size_t)(bb * 16 + hh) * 1024 + (size_t)(s >> 1)) * 128 + e * 2 + (s & 1)] = val;
      }
    }
  }
}

// ---------------------------------------------------------------------------
// Fused attention: per block = (b, h, 16 query rows)
//   scores (masked-skip) -> LDS strip -> softmax -> pattern write -> z = P @ V
// vpair is the k-paired V layout written by the V projection (omode 1).
// ---------------------------------------------------------------------------
#define ATT_S 2048
#define ATT_H 16
#define ATT_D 64
#define LDP   2052   // padded score-row stride: lane bank = 4*m, conflict-free
#define LDQ   68     // padded Q stride

__global__ __launch_bounds__(256) void attn_softmax_wmma(
    const float* __restrict__ q, const float* __restrict__ kmat,
    const float* __restrict__ vpair, float* __restrict__ pattern,
    float* __restrict__ z)
{
  __shared__ float Plds[16 * LDP];   // 131,328 B score/prob strip
  __shared__ float Scr[16 * LDQ];    // Q staging, later split-K scratch

  const int t    = threadIdx.x;
  const int wave = t >> 5;
  const int lane = t & 31;
  const int ml   = lane & 15;
  const int hi   = lane >> 4;

  const int qt = blockIdx.x & 127;
  const int h  = (blockIdx.x >> 7) & 15;
  const int b  = blockIdx.x >> 11;
  const int q0 = qt << 4;

  const size_t rowsz = (size_t)ATT_H * ATT_D;                       // 1024
  const size_t base  = ((size_t)b * ATT_S) * rowsz + (size_t)h * ATT_D;

  // ---- stage Q * (1/sqrt(64)) ----
  {
    const int linear = t << 2;           // 16*64 floats, 1 float4/thread
    const int r = linear >> 6, e = linear & 63;
    const float4 qv = *(const float4*)(q + base + (size_t)(q0 + r) * rowsz + e);
    float* d = &Scr[r * LDQ + e];
    d[0] = qv.x * 0.125f; d[1] = qv.y * 0.125f;
    d[2] = qv.z * 0.125f; d[3] = qv.w * 0.125f;
  }
  __syncthreads();

  // ---- scores: 16x2048 = Q @ K^T, skip fully-masked tiles ----
  for (int nt = wave; nt < 128; nt += 8) {
    const int n0 = nt << 4;
    if (n0 > q0 + 15) break;             // causal: everything further is masked
    v8f acc = {};
    #pragma unroll
    for (int kk = 0; kk < ATT_D; kk += 4) {
      const int kb = kk + hi * 2;
      const v2f a  = *(const v2f*)&Scr[ml * LDQ + kb];
      const v2f bk = *(const v2f*)(kmat + base + (size_t)(n0 + ml) * rowsz + kb);
      acc = WMMA_F32(a, bk, acc);
    }
    #pragma unroll
    for (int r = 0; r < 8; r++)
      Plds[(r + hi * 8) * LDP + n0 + ml] = acc[r];
  }
  __syncthreads();

  // ---- softmax per row (2 rows per wave), write pattern ----
  #pragma unroll
  for (int rr = 0; rr < 2; rr++) {
    const int r  = wave + rr * 8;
    const int qg = q0 + r;               // last unmasked key position
    float mx = -1e30f;
    for (int c = lane; c <= qg; c += 32) mx = fmaxf(mx, Plds[r * LDP + c]);
    #pragma unroll
    for (int off = 16; off > 0; off >>= 1) mx = fmaxf(mx, __shfl_xor(mx, off, 32));
    float sum = 0.f;
    for (int c = lane; c <= qg; c += 32) {
      const float p = __expf(Plds[r * LDP + c] - mx);
      Plds[r * LDP + c] = p;
      sum += p;
    }
    #pragma unroll
    for (int off = 16; off > 0; off >>= 1) sum += __shfl_xor(sum, off, 32);
    const float inv = 1.0f / sum;
    const size_t prow = ((((size_t)b * ATT_H + h) * ATT_S) + qg) * (size_t)ATT_S;
    for (int c = lane; c < ATT_S; c += 32) {
      const float p = (c <= qg) ? Plds[r * LDP + c] * inv : 0.f;
      Plds[r * LDP + c] = p;             // also feeds z phase
      pattern[prow + c] = p;
    }
  }
  __syncthreads();

  // ---- z = P @ V, split-K across wave pairs, K truncated at diagonal ----
  const int ns    = wave & 3;            // 16-wide slice of d_head
  const int khalf = (wave >> 2) << 10;   // 0 or 1024
  const size_t vpb = ((size_t)b * ATT_H + h) * (size_t)(1024 * 128) +
                     (size_t)(ns * 16 + ml) * 2;
  v8f acc = {};
  int kend = (q0 + 16) - khalf;
  if (kend > 1024) kend = 1024;
  for (int kk = 0; kk < kend; kk += 4) {
    const int kb = khalf + kk + hi * 2;  // even
    const v2f a  = *(const v2f*)&Plds[ml * LDP + kb];
    const v2f bv = *(const v2f*)(vpair + vpb + (size_t)(kb >> 1) * 128);
    acc = WMMA_F32(a, bv, acc);
  }
  if (wave >= 4) {
    #pragma unroll
    for (int r = 0; r < 8; r++)
      Scr[(wave - 4) * 256 + r * 32 + lane] = acc[r];
  }
  __syncthreads();
  if (wave < 4) {
    #pragma unroll
    for (int r = 0; r < 8; r++) {
      const float val = acc[r] + Scr[wave * 256 + r * 32 + lane];
      const int qr = q0 + r + hi * 8;
      z[base + (size_t)qr * rowsz + ns * 16 + ml] = val;   // z[b][q][h][e]
    }
  }
}

// ---------------------------------------------------------------------------
extern "C" void kernel_launch(void* const* d_in, const int* in_sizes, int n_in,
                              void* d_out, int out_size, void* d_ws, size_t ws_size,
                              hipStream_t stream) {
  (void)in_sizes; (void)n_in; (void)out_size; (void)ws_size;
  const float* x   = (const float*)d_in[0];
  const float* W_Q = (const float*)d_in[1];
  const float* b_Q = (const float*)d_in[2];
  const float* W_K = (const float*)d_in[3];
  const float* b_K = (const float*)d_in[4];
  const float* W_V = (const float*)d_in[5];
  const float* b_V = (const float*)d_in[6];
  const float* W_O = (const float*)d_in[7];
  const float* b_O = (const float*)d_in[8];

  const size_t QKV = (size_t)4 * 2048 * 1024;   // 8,388,608 elements

  float* attn_out = (float*)d_out;              // [B,S,1024]
  float* pattern  = attn_out + QKV;             // [B,H,S,S]
  float* ws = (float*)d_ws;
  float* qb = ws;
  float* kb = ws + QKV;
  float* vb = ws + 2 * QKV;                     // k-paired V layout
  float* zb = ws + 3 * QKV;

  const int M = 8192, N = 1024, K = 1024;
  dim3 blk(256);
  dim3 grid((M / 128) * (N / 64));              // 1024 blocks

  gemm_wmma_f32<<<grid, blk, 0, stream>>>(x,  W_Q, b_Q, qb, M, N, K, 1, 0);
  gemm_wmma_f32<<<grid, blk, 0, stream>>>(x,  W_K, b_K, kb, M, N, K, 1, 0);
  gemm_wmma_f32<<<grid, blk, 0, stream>>>(x,  W_V, b_V, vb, M, N, K, 1, 1);
  attn_softmax_wmma<<<dim3(4 * 16 * 128), blk, 0, stream>>>(qb, kb, vb, pattern, zb);
  gemm_wmma_f32<<<grid, blk, 0, stream>>>(zb, W_O, b_O, attn_out, M, N, K, 0, 0);
}